// InterAgg_53266184405178
// MI455X (gfx1250) — compile-verified
//
#include <hip/hip_runtime.h>

// CARE-GNN threshold_inter_agg, fused for MI455X (gfx1250).
//   combined = relu((self + t0*n0 + t1*n1 + t2*n2) @ W)   [n,128]
//   scores   = self @ clf_w + clf_b                        [n,2]
// Linear fusion cuts GEMM FLOPs 4x; kernel is HBM-bound (~462 MB -> ~20us
// @ 23.3 TB/s). fp32 precision kept via V_WMMA_F32_16X16X4_F32.
// v2: K-pair-interleaved W in LDS (single b64 B-fragment, no v_mov shuffles,
//     conflict-free banks) + register-pipelined global streams with
//     double-buffered E tile (hides HBM latency behind WMMA).

typedef __attribute__((ext_vector_type(2))) float v2f;
typedef __attribute__((ext_vector_type(4))) float v4f;
typedef __attribute__((ext_vector_type(8))) float v8f;

#define FEAT    256
#define EMB     128
#define ROWS_WG 128          // 8 waves x 16 rows
#define KCHUNK  32
#define EPITCH  36           // E row pitch: A-frag b64 -> all 64 banks distinct
#define WP2     288          // W k-pair pitch (== 32 mod 64): B-frag b64 conflict-free
#define NPAIRS  (FEAT / 2)
#define LDSF_W  (NPAIRS * WP2)          // 36864 floats
#define LDSF_E  (ROWS_WG * EPITCH)      // 4608 floats per buffer
#define LDSF_CW (FEAT * 2)              // 512 floats
#define LDS_FLOATS (LDSF_W + 2 * LDSF_E + LDSF_CW)   // 186368 bytes

__global__ __launch_bounds__(256)
void care_interagg_wmma(const float* __restrict__ self_feats,
                        const float* __restrict__ neigh_feats,
                        const float* __restrict__ weight,
                        const float* __restrict__ clf_w,
                        const float* __restrict__ clf_b,
                        const float* __restrict__ thresholds,
                        float* __restrict__ out_combined,
                        float* __restrict__ out_scores,
                        int n)
{
    extern __shared__ float smem[];
    float* sWp = smem;                         // [128 pairs][WP2]
    float* sE  = smem + LDSF_W;                // [2][ROWS_WG][EPITCH]
    float* sCW = smem + LDSF_W + 2 * LDSF_E;   // [256][2]

    const int t    = threadIdx.x;              // 0..255
    const int wg   = blockIdx.x;
    const int lane = t & 31;
    const int wave = t >> 5;                   // 0..7
    const int hl   = lane >> 4;                // half-wave: K offset +2
    const int l16  = lane & 15;

    const float t0 = thresholds[0];
    const float t1 = thresholds[1];
    const float t2 = thresholds[2];

    // ---- stage W into LDS, K-pair interleaved: sWp[kp][2c + (k&1)] ----
    #pragma unroll 4
    for (int j = 0; j < 16; ++j) {
        int u  = t + 256 * j;                  // pair-unit index
        int kp = u >> 5;                       // 0..127
        int c  = (u & 31) << 2;                // 0..124
        v4f w0 = *(const v4f*)(weight + (size_t)(2 * kp)     * EMB + c);
        v4f w1 = *(const v4f*)(weight + (size_t)(2 * kp + 1) * EMB + c);
        float* d = sWp + (size_t)kp * WP2 + 2 * c;
        *(v4f*)(d)     = (v4f){w0.x, w1.x, w0.y, w1.y};
        *(v4f*)(d + 4) = (v4f){w0.z, w1.z, w0.w, w1.w};
    }
    if (t < 128) *(v4f*)(sCW + t * 4) = *(const v4f*)(clf_w + t * 4);

    // accumulators: 8 N-tiles of 16x16 per wave
    v8f acc[8];
    #pragma unroll
    for (int i = 0; i < 8; ++i) acc[i] = (v8f){0.f,0.f,0.f,0.f,0.f,0.f,0.f,0.f};

    float sc0[4], sc1[4];
    #pragma unroll
    for (int i = 0; i < 4; ++i) { sc0[i] = 0.f; sc1[i] = 0.f; }

    const int row0 = wg * ROWS_WG;
    const int rloc = t >> 3;                   // 0..31 (+32*i)
    const int c4s  = t & 7;                    // staging float4 column

    // pipeline registers: raw streams for the next chunk (4 units/thread)
    v4f ps[4], p0[4], p1[4], p2[4];
    const size_t nF = (size_t)n * FEAT;

    auto issue_loads = [&](int kc) {
        #pragma unroll
        for (int i = 0; i < 4; ++i) {
            int r    = rloc + 32 * i;                       // 0..127
            int grow = row0 + r; if (grow > n - 1) grow = n - 1;
            int col  = kc * KCHUNK + (c4s << 2);
            const float* sp = self_feats  + (size_t)grow * FEAT + col;
            const float* np = neigh_feats + (size_t)grow * FEAT + col;
            ps[i] = __builtin_nontemporal_load((const v4f*)sp);
            p0[i] = __builtin_nontemporal_load((const v4f*)np);
            p1[i] = __builtin_nontemporal_load((const v4f*)(np + nF));
            p2[i] = __builtin_nontemporal_load((const v4f*)(np + 2 * nF));
        }
    };

    issue_loads(0);

    const int arow = wave * 16 + l16;          // local row for A fragment

    for (int kc = 0; kc < FEAT / KCHUNK; ++kc) {
        // ---- fuse pipelined regs -> E buffer (kc&1), fold scores ----
        float* buf = sE + (size_t)(kc & 1) * LDSF_E;
        #pragma unroll
        for (int i = 0; i < 4; ++i) {
            int r   = rloc + 32 * i;
            int col = kc * KCHUNK + (c4s << 2);
            v4f s  = ps[i];
            v4f ev = s + t0 * p0[i] + t1 * p1[i] + t2 * p2[i];
            *(v4f*)(buf + r * EPITCH + (c4s << 2)) = ev;
            sc0[i] += s.x * sCW[(col + 0) * 2 + 0] + s.y * sCW[(col + 1) * 2 + 0]
                    + s.z * sCW[(col + 2) * 2 + 0] + s.w * sCW[(col + 3) * 2 + 0];
            sc1[i] += s.x * sCW[(col + 0) * 2 + 1] + s.y * sCW[(col + 1) * 2 + 1]
                    + s.z * sCW[(col + 2) * 2 + 1] + s.w * sCW[(col + 3) * 2 + 1];
        }
        // ---- prefetch next chunk (waited at next fuse, hidden by WMMA) ----
        if (kc + 1 < FEAT / KCHUNK) issue_loads(kc + 1);
        __syncthreads();

        // ---- WMMA: 8 K-steps of 4, 8 N-tiles ----
        const float* eb = buf + arow * EPITCH;
        #pragma unroll
        for (int kk = 0; kk < KCHUNK / 4; ++kk) {
            const int kofs = (kk << 2) + (hl << 1);
            v2f a = *(const v2f*)(eb + kofs);                  // {K, K+1}
            const float* wp = sWp
                + (size_t)(kc * (KCHUNK / 2) + kk * 2 + hl) * WP2 + 2 * l16;
            v2f b[8];
            #pragma unroll
            for (int nt = 0; nt < 8; ++nt)
                b[nt] = *(const v2f*)(wp + nt * 32);           // {W[k][c], W[k+1][c]}
            #pragma unroll
            for (int nt = 0; nt < 8; ++nt)
                acc[nt] = __builtin_amdgcn_wmma_f32_16x16x4_f32(
                    false, a, false, b[nt], (short)0, acc[nt], false, false);
        }
    }

    // ---- relu + store combined ----
    {
        const int m0 = row0 + wave * 16;
        #pragma unroll
        for (int nt = 0; nt < 8; ++nt) {
            const int coln = nt * 16 + l16;
            #pragma unroll
            for (int g = 0; g < 8; ++g) {
                int row = m0 + g + hl * 8;      // C layout: VGPR g -> M=g / M=g+8
                if (row < n) {
                    float v = acc[nt][g];
                    __builtin_nontemporal_store(v > 0.f ? v : 0.f,
                        out_combined + (size_t)row * EMB + coln);
                }
            }
        }
    }

    // ---- reduce + store scores (8-lane groups share a row set) ----
    {
        const float b0 = clf_b[0], b1 = clf_b[1];
        #pragma unroll
        for (int i = 0; i < 4; ++i) {
            #pragma unroll
            for (int m = 4; m >= 1; m >>= 1) {
                sc0[i] += __shfl_xor(sc0[i], m, 32);
                sc1[i] += __shfl_xor(sc1[i], m, 32);
            }
        }
        if (c4s == 0) {
            #pragma unroll
            for (int i = 0; i < 4; ++i) {
                int row = row0 + rloc + 32 * i;
                if (row < n) {
                    __builtin_nontemporal_store(sc0[i] + b0, out_scores + (size_t)row * 2 + 0);
                    __builtin_nontemporal_store(sc1[i] + b1, out_scores + (size_t)row * 2 + 1);
                }
            }
        }
    }
}

extern "C" void kernel_launch(void* const* d_in, const int* in_sizes, int n_in,
                              void* d_out, int out_size, void* d_ws, size_t ws_size,
                              hipStream_t stream) {
    const float* self_feats  = (const float*)d_in[0];
    const float* neigh_feats = (const float*)d_in[1];
    const float* weight      = (const float*)d_in[2];
    const float* clf_w       = (const float*)d_in[3];
    const float* clf_b       = (const float*)d_in[4];
    const float* thresholds  = (const float*)d_in[5];

    const int n = in_sizes[0] / FEAT;
    float* out_combined = (float*)d_out;
    float* out_scores   = (float*)d_out + (size_t)n * EMB;

    const int grid = (n + ROWS_WG - 1) / ROWS_WG;
    const size_t lds_bytes = (size_t)LDS_FLOATS * sizeof(float); // ~182 KB

    care_interagg_wmma<<<grid, 256, lds_bytes, stream>>>(
        self_feats, neigh_feats, weight, clf_w, clf_b, thresholds,
        out_combined, out_scores, n);
}